// Partition_enhanced_GCN_31482110280434
// MI455X (gfx1250) — compile-verified
//
#include <hip/hip_runtime.h>
#include <hip/hip_bf16.h>

typedef float v2f __attribute__((ext_vector_type(2)));
typedef float v8f __attribute__((ext_vector_type(8)));

#define HID 128
#define NGRAPHS 64
#define NCLUST 4
#define NLAYERS 3

// ---------------------------------------------------------------------------
// degree / normalization
// ---------------------------------------------------------------------------
__global__ void deg_init_kernel(float* __restrict__ deg, int n) {
    int i = blockIdx.x * blockDim.x + threadIdx.x;
    if (i < n) deg[i] = 1.0f;  // self loop
}

__global__ void deg_edges_kernel(const int* __restrict__ col, float* __restrict__ deg, int e) {
    int i = blockIdx.x * blockDim.x + threadIdx.x;
    if (i < e) atomicAdd(&deg[col[i]], 1.0f);
}

__global__ void dinv_kernel(const float* __restrict__ deg, float* __restrict__ dinv, int n) {
    int i = blockIdx.x * blockDim.x + threadIdx.x;
    if (i < n) {
        float d = deg[i];
        dinv[i] = (d > 0.0f) ? rsqrtf(d) : 0.0f;
    }
}

__global__ void zero_kernel(float* __restrict__ p, int n) {
    int i = blockIdx.x * blockDim.x + threadIdx.x;
    if (i < n) p[i] = 0.0f;
}

// ---------------------------------------------------------------------------
// aggregation: m = A_norm @ h   (self-loop init, then edge scatter-add)
// ---------------------------------------------------------------------------
template <int CH>
__global__ void agg_init_kernel(const float* __restrict__ h, const float* __restrict__ dinv,
                                float* __restrict__ m, int n) {
    int i = blockIdx.x;
    int ch = threadIdx.x;
    if (i < n) {
        float d = dinv[i];
        m[(size_t)i * CH + ch] = d * d * h[(size_t)i * CH + ch];
    }
}

template <int CH>
__global__ void agg_edges_kernel(const float* __restrict__ h, const int* __restrict__ row,
                                 const int* __restrict__ col, const float* __restrict__ dinv,
                                 float* __restrict__ m, int e) {
    int idx = blockIdx.x;
    if (idx >= e) return;
    int r = row[idx];
    int c = col[idx];
    float w = dinv[r] * dinv[c];
    int ch = threadIdx.x;
    atomicAdd(&m[(size_t)c * CH + ch], w * h[(size_t)r * CH + ch]);
}

// ---------------------------------------------------------------------------
// per-node GEMM with cluster-selected weights using V_WMMA_F32_16X16X4_F32
//   out[i] = m[i] @ W[cluster[i]] + b[cluster[i]]
// One wave (32 threads) per 16-node tile.
// A layout (16x4 f32): lane l<16 / l>=16, frag v -> A[M = l&15][K = 4k + 2*half + v]
// B layout (4x16 f32): frag v -> B[K = 4k + 2*half + v][N = nb + (l&15)]
// C/D layout (16x16 f32): frag v, lane -> D[M = v + 8*half][N = nb + (l&15)]
// ---------------------------------------------------------------------------
template <int CH>
__global__ __launch_bounds__(32) void gcn_gemm_wmma_kernel(
    const float* __restrict__ m,        // [n][CH]
    const float* __restrict__ W,        // [NCLUST][CH][HID]
    const float* __restrict__ bias,     // [NCLUST][HID]
    const int* __restrict__ cluster,    // [n]
    float* __restrict__ out,            // [n][HID]
    int n) {
    const int lane = threadIdx.x;
    const int half = lane >> 4;
    const int l = lane & 15;
    const int tb = blockIdx.x * 16;

    // Preload A fragments for this tile (same for all clusters / column tiles).
    // 8-byte aligned: (tb+l)*CH + 4k + 2*half is always even.
    v2f afrag[CH / 4];
    const float* mrow = m + (size_t)(tb + l) * CH;
#pragma unroll
    for (int k = 0; k < CH / 4; ++k) {
        afrag[k] = *(const v2f*)(mrow + 4 * k + 2 * half);
    }

    // Preload cluster ids of my 8 output rows (hoisted out of all loops).
    int myclust[8];
    bool myvalid[8];
#pragma unroll
    for (int v = 0; v < 8; ++v) {
        int node = tb + v + 8 * half;
        myvalid[v] = (node < n);
        myclust[v] = myvalid[v] ? cluster[node] : -1;
    }

    for (int c = 0; c < NCLUST; ++c) {
        // wave-uniform skip if this cluster has no rows in the tile
        bool anyrow = false;
#pragma unroll
        for (int v = 0; v < 8; ++v) anyrow |= (myclust[v] == c);
        if (__ballot(anyrow) == 0ull) continue;

        const float* Wc = W + (size_t)c * CH * HID;
        const float* bc = bias + c * HID;
        for (int nb = 0; nb < HID; nb += 16) {
            v8f acc = {};
#pragma unroll
            for (int k = 0; k < CH / 4; ++k) {
                const float* wk = Wc + (size_t)(4 * k + 2 * half) * HID + nb + l;
                v2f bfrag;
                bfrag.x = wk[0];
                bfrag.y = wk[HID];
                acc = __builtin_amdgcn_wmma_f32_16x16x4_f32(
                    false, afrag[k], false, bfrag, (short)0, acc, false, false);
            }
            float bval = bc[nb + l];
#pragma unroll
            for (int v = 0; v < 8; ++v) {
                if (myclust[v] == c) {
                    int node = tb + v + 8 * half;
                    out[(size_t)node * HID + nb + l] = acc[v] + bval;
                }
            }
        }
    }
}

// ---------------------------------------------------------------------------
// graph pooling: zpre[batch[i], t*HID + ch] += h[i][ch]
// batch is sorted -> run-length accumulate per thread, flush on graph change.
// Block handles 64 nodes, one thread per channel.
// ---------------------------------------------------------------------------
__global__ void pool_kernel(const float* __restrict__ h, const int* __restrict__ batch,
                            float* __restrict__ zpre, int n, int t) {
    int ch = threadIdx.x;
    int n0 = blockIdx.x * 64;
    int n1 = min(n0 + 64, n);
    float acc = 0.0f;
    int curg = -1;
    for (int i = n0; i < n1; ++i) {
        int g = batch[i];
        if (g != curg) {
            if (curg >= 0)
                atomicAdd(&zpre[(size_t)curg * (HID * NLAYERS) + t * HID + ch], acc);
            curg = g;
            acc = 0.0f;
        }
        acc += h[(size_t)i * HID + ch];
    }
    if (curg >= 0)
        atomicAdd(&zpre[(size_t)curg * (HID * NLAYERS) + t * HID + ch], acc);
}

// ---------------------------------------------------------------------------
// MLP head
// ---------------------------------------------------------------------------
__global__ void mlp1_kernel(const float* __restrict__ zpre, const float* __restrict__ W1,
                            const float* __restrict__ b1, float* __restrict__ z1) {
    int g = blockIdx.x;   // 0..63
    int j = threadIdx.x;  // 0..127
    float s = b1[j];
    const float* zg = zpre + (size_t)g * (HID * NLAYERS);
    for (int k = 0; k < HID * NLAYERS; ++k) s += zg[k] * W1[(size_t)k * HID + j];
    z1[(size_t)g * HID + j] = s;
}

__global__ void bn_relu_kernel(const float* __restrict__ z1, const float* __restrict__ gamma,
                               const float* __restrict__ beta, float* __restrict__ z2) {
    int j = threadIdx.x;  // 0..127, single block
    float vals[NGRAPHS];
    float mu = 0.0f;
    for (int g = 0; g < NGRAPHS; ++g) {
        vals[g] = z1[(size_t)g * HID + j];
        mu += vals[g];
    }
    mu *= (1.0f / NGRAPHS);
    float var = 0.0f;
    for (int g = 0; g < NGRAPHS; ++g) {
        float d = vals[g] - mu;
        var += d * d;
    }
    var *= (1.0f / NGRAPHS);
    float s = rsqrtf(var + 1e-5f) * gamma[j];
    float bb = beta[j];
    for (int g = 0; g < NGRAPHS; ++g) {
        float v = (vals[g] - mu) * s + bb;
        z2[(size_t)g * HID + j] = fmaxf(v, 0.0f);
    }
}

__global__ void mlp2_kernel(const float* __restrict__ z2, const float* __restrict__ W2,
                            const float* __restrict__ b2, float* __restrict__ out) {
    int g = blockIdx.x;   // 0..63
    int o = threadIdx.x;  // 0..15
    float s = b2[o];
    const float* zg = z2 + (size_t)g * HID;
    for (int j = 0; j < HID; ++j) s += zg[j] * W2[(size_t)j * 16 + o];
    out[(size_t)g * 16 + o] = s;
}

// ---------------------------------------------------------------------------
// launch
// ---------------------------------------------------------------------------
extern "C" void kernel_launch(void* const* d_in, const int* in_sizes, int n_in,
                              void* d_out, int out_size, void* d_ws, size_t ws_size,
                              hipStream_t stream) {
    const float* x = (const float*)d_in[0];         // [N][64]
    const int* cluster = (const int*)d_in[1];       // [N]
    const int* ei = (const int*)d_in[2];            // [2][E]
    const int* batch = (const int*)d_in[3];         // [N]
    const float* W0 = (const float*)d_in[4];        // [4][64][128]
    const float* b0 = (const float*)d_in[5];        // [4][128]
    const float* Wh = (const float*)d_in[6];        // [2][4][128][128]
    const float* bh = (const float*)d_in[7];        // [2][4][128]
    const float* W1 = (const float*)d_in[8];        // [384][128]
    const float* b1 = (const float*)d_in[9];        // [128]
    const float* gamma = (const float*)d_in[10];    // [128]
    const float* beta = (const float*)d_in[11];     // [128]
    const float* W2 = (const float*)d_in[12];       // [128][16]
    const float* b2 = (const float*)d_in[13];       // [16]

    const int n = in_sizes[1];       // N (cluster count)
    const int e = in_sizes[2] / 2;   // E
    const int* erow = ei;            // source
    const int* ecol = ei + e;        // target

    // workspace layout (floats)
    float* ws = (float*)d_ws;
    float* bufA = ws;                               // [N][128]  layer outputs h
    float* bufB = bufA + (size_t)n * HID;           // [N][128]  aggregated m
    float* deg = bufB + (size_t)n * HID;            // [N]
    float* dinv = deg + n;                          // [N]
    float* zpre = dinv + n;                         // [64][384]
    float* z1 = zpre + NGRAPHS * HID * NLAYERS;     // [64][128]
    float* z2 = z1 + NGRAPHS * HID;                 // [64][128]
    float* out = (float*)d_out;                     // [64][16]

    // --- gcn_norm: deg over targets incl. self loops, then rsqrt ---
    deg_init_kernel<<<(n + 255) / 256, 256, 0, stream>>>(deg, n);
    deg_edges_kernel<<<(e + 255) / 256, 256, 0, stream>>>(ecol, deg, e);
    dinv_kernel<<<(n + 255) / 256, 256, 0, stream>>>(deg, dinv, n);
    zero_kernel<<<(NGRAPHS * HID * NLAYERS + 255) / 256, 256, 0, stream>>>(
        zpre, NGRAPHS * HID * NLAYERS);

    const int gemm_blocks = (n + 15) / 16;

    // --- layer 0: CH = 64 input ---
    agg_init_kernel<64><<<n, 64, 0, stream>>>(x, dinv, bufB, n);
    agg_edges_kernel<64><<<e, 64, 0, stream>>>(x, erow, ecol, dinv, bufB, e);
    gcn_gemm_wmma_kernel<64><<<gemm_blocks, 32, 0, stream>>>(bufB, W0, b0, cluster, bufA, n);
    pool_kernel<<<(n + 63) / 64, HID, 0, stream>>>(bufA, batch, zpre, n, 0);

    // --- layers 1..2: CH = 128 ---
    for (int t = 1; t < NLAYERS; ++t) {
        const float* Wt = Wh + (size_t)(t - 1) * NCLUST * HID * HID;
        const float* bt = bh + (size_t)(t - 1) * NCLUST * HID;
        agg_init_kernel<128><<<n, 128, 0, stream>>>(bufA, dinv, bufB, n);
        agg_edges_kernel<128><<<e, 128, 0, stream>>>(bufA, erow, ecol, dinv, bufB, e);
        gcn_gemm_wmma_kernel<128><<<gemm_blocks, 32, 0, stream>>>(bufB, Wt, bt, cluster, bufA, n);
        pool_kernel<<<(n + 63) / 64, HID, 0, stream>>>(bufA, batch, zpre, n, t);
    }

    // --- MLP head + batch norm + relu ---
    mlp1_kernel<<<NGRAPHS, HID, 0, stream>>>(zpre, W1, b1, z1);
    bn_relu_kernel<<<1, HID, 0, stream>>>(z1, gamma, beta, z2);
    mlp2_kernel<<<NGRAPHS, 16, 0, stream>>>(z2, W2, b2, out);
}